// HiddenHarsanyiBlock_515396075882
// MI455X (gfx1250) — compile-verified
//
#include <hip/hip_runtime.h>
#include <math.h>

// CDNA5 wave32 WMMA vector types
typedef __attribute__((ext_vector_type(2))) float v2f;   // 16x4 / 4x16 f32 fragment: 2 VGPRs
typedef __attribute__((ext_vector_type(8))) float v8f;   // 16x16 f32 C/D fragment: 8 VGPRs

#define B_DIM 256
#define I_DIM 1024
#define O_DIM 1024
#define TILES_N (O_DIM / 16)          // 64
#define TILES_M (B_DIM / 16)          // 16
#define NT 2                          // column tiles per wave (share A fragment)
#define WAVES_PER_BLOCK 8

// Branch-free L2(x) = log2(tanh(100*|x|)) clamped to -1e4, using raw HW
// transcendentals only (v_exp_f32, v_rcp_f32, v_log_f32) -- no IEEE div
// ladder, no libm range-reduction branches.
//   u = 2^(-200*log2(e)*|x|) = e^(-200|x|)
//   t = (1-u) * rcp(1+u) = tanh(100|x|)
// Working in the log2 domain: delta = prod(t) = 2^(sum log2 t), so the
// epilogue is a single v_exp_f32.
__device__ __forceinline__ float log2_tanh_gamma(float xv) {
    const float C = -288.53900817779268f;                 // -200 * log2(e)
    const float u = __builtin_amdgcn_exp2f(C * fabsf(xv));
    const float t = (1.0f - u) * __builtin_amdgcn_rcpf(1.0f + u);
    return fmaxf(__builtin_amdgcn_logf(t), -1.0e4f);      // clamp kills -inf at x==0
}

// Fused kernel: one wave per 16x(16*NT) output strip.
//   accY[j] = x_tile(16xK) @ (fc*mask)^T_tile_j(Kx16)     via V_WMMA_F32_16X16X4_F32
//   accD[j] = L2_tile(16xK) @ mask^T_tile_j(Kx16)
//   delta = exp2(accD); y = relu(accY * delta)
__global__ __launch_bounds__(32 * WAVES_PER_BLOCK)
void harsanyi_fused_wmma(const float* __restrict__ x,
                         const float* __restrict__ vw,
                         const float* __restrict__ fcw,
                         float* __restrict__ out) {
    const int lane  = threadIdx.x & 31;
    const int wave  = threadIdx.x >> 5;
    const int wid   = blockIdx.x * WAVES_PER_BLOCK + wave;   // 0..511
    const int strips_n = TILES_N / NT;                        // 32
    const int tile_m = wid / strips_n;                        // 0..15
    const int strip  = wid % strips_n;                        // 0..31
    const int row0 = tile_m * 16;                             // batch-row base
    const int col0 = strip * (16 * NT);                       // output-neuron base
    const int half = lane >> 4;         // 0: K pair {0,1}; 1: K pair {2,3}
    const int lm   = lane & 15;         // M index (A side) / N index (B side)

    // A side: this lane supplies row (row0+lm) of x / L2.
    // B side: this lane supplies K-data of neuron (col0 + j*16 + lm) for tile j.
    const float* __restrict__ xrow  = x   + (size_t)(row0 + lm) * I_DIM;
    const float* __restrict__ vrow0 = vw  + (size_t)(col0 + lm) * I_DIM;
    const float* __restrict__ frow0 = fcw + (size_t)(col0 + lm) * I_DIM;
    const float* __restrict__ vrow1 = vrow0 + (size_t)16 * I_DIM;
    const float* __restrict__ frow1 = frow0 + (size_t)16 * I_DIM;

    v8f accY0 = {}, accY1 = {};
    v8f accD0 = {}, accD1 = {};

    for (int k = 0; k < I_DIM; k += 4) {
        const int kk = k + 2 * half;    // per-lane K offset inside the 16x4 fragment

        v2f xa  = *(const v2f*)(xrow  + kk);   // A fragment of x (shared by both tiles)
        v2f vv0 = *(const v2f*)(vrow0 + kk);
        v2f ff0 = *(const v2f*)(frow0 + kk);
        v2f vv1 = *(const v2f*)(vrow1 + kk);
        v2f ff1 = *(const v2f*)(frow1 + kk);

        // L2 fragment (computed once, feeds both delta-GEMMs)
        v2f la;
        la.x = log2_tanh_gamma(xa.x);
        la.y = log2_tanh_gamma(xa.y);

        // Masked B fragments as direct selects (one v_cndmask each)
        v2f bw0, bm0, bw1, bm1;
        bw0.x = vv0.x > 0.0f ? ff0.x : 0.0f;
        bw0.y = vv0.y > 0.0f ? ff0.y : 0.0f;
        bm0.x = vv0.x > 0.0f ? 1.0f : 0.0f;
        bm0.y = vv0.y > 0.0f ? 1.0f : 0.0f;
        bw1.x = vv1.x > 0.0f ? ff1.x : 0.0f;
        bw1.y = vv1.y > 0.0f ? ff1.y : 0.0f;
        bm1.x = vv1.x > 0.0f ? 1.0f : 0.0f;
        bm1.y = vv1.y > 0.0f ? 1.0f : 0.0f;

        accY0 = __builtin_amdgcn_wmma_f32_16x16x4_f32(false, xa, false, bw0,
                                                      (short)0, accY0, false, false);
        accD0 = __builtin_amdgcn_wmma_f32_16x16x4_f32(false, la, false, bm0,
                                                      (short)0, accD0, false, false);
        accY1 = __builtin_amdgcn_wmma_f32_16x16x4_f32(false, xa, false, bw1,
                                                      (short)0, accY1, false, false);
        accD1 = __builtin_amdgcn_wmma_f32_16x16x4_f32(false, la, false, bm1,
                                                      (short)0, accD1, false, false);
    }

    // C/D layout: VGPR r, lanes 0-15 -> M=r, lanes 16-31 -> M=8+r; N = lane&15.
    const int orow0 = row0 + 8 * half;
    float* __restrict__ yout = out;
    float* __restrict__ dout = out + (size_t)B_DIM * O_DIM;

#pragma unroll
    for (int r = 0; r < 8; ++r) {
        const size_t rowoff = (size_t)(orow0 + r) * O_DIM;

        const float d0 = __builtin_amdgcn_exp2f(accD0[r]);   // underflow->0 when suppressed
        float y0 = accY0[r] * d0;
        y0 = y0 > 0.0f ? y0 : 0.0f;                          // relu
        const size_t i0 = rowoff + (size_t)(col0 + lm);
        yout[i0] = y0;
        dout[i0] = d0;

        const float d1 = __builtin_amdgcn_exp2f(accD1[r]);
        float y1 = accY1[r] * d1;
        y1 = y1 > 0.0f ? y1 : 0.0f;
        const size_t i1 = rowoff + (size_t)(col0 + 16 + lm);
        yout[i1] = y1;
        dout[i1] = d1;
    }
}

extern "C" void kernel_launch(void* const* d_in, const int* in_sizes, int n_in,
                              void* d_out, int out_size, void* d_ws, size_t ws_size,
                              hipStream_t stream) {
    (void)in_sizes; (void)n_in; (void)out_size; (void)d_ws; (void)ws_size;
    const float* x   = (const float*)d_in[0];
    const float* vw  = (const float*)d_in[1];
    const float* fcw = (const float*)d_in[2];
    float* out = (float*)d_out;

    const int total_waves = TILES_M * (TILES_N / NT);          // 512
    dim3 grid(total_waves / WAVES_PER_BLOCK);                  // 64 blocks
    dim3 block(32 * WAVES_PER_BLOCK);                          // 256 threads = 8 wave32
    hipLaunchKernelGGL(harsanyi_fused_wmma, grid, block, 0, stream, x, vw, fcw, out);
}